// ScaledDotProductRelativeAttention_59734405153252
// MI455X (gfx1250) — compile-verified
//
#include <hip/hip_runtime.h>
#include <hip/hip_bf16.h>

#if defined(__HIP_DEVICE_COMPILE__)
#if !__has_builtin(__builtin_amdgcn_wmma_f32_16x16x4_f32)
#error "wmma f32 16x16x4 builtin missing on this toolchain (device pass)"
#endif
#endif

typedef float v2f __attribute__((ext_vector_type(2)));
typedef float v8f __attribute__((ext_vector_type(8)));

#define WMMA_F32(A, B, C) \
  __builtin_amdgcn_wmma_f32_16x16x4_f32(false, (A), false, (B), (short)0, (C), false, false)

// Fragment where the LDS tile is stored [row][K]: row = lane&15, contiguous K pair.
// Matches ISA 32-bit A 16x4 layout AND B 4x16 layout when B-source is [N][K] (i.e. K^T).
__device__ __forceinline__ v2f ldfrag_rowmajor(const float* base, int pitch, int lane, int k0) {
  int r  = lane & 15;
  int kk = (lane >> 4) << 1;
  return *(const v2f*)(base + r * pitch + k0 + kk);
}

// Fragment where the LDS tile is stored [K][N] (dense V tile): two strided loads.
__device__ __forceinline__ v2f ldfrag_colmajor(const float* base, int pitch, int lane, int k0) {
  int n  = lane & 15;
  int kk = (lane >> 4) << 1;
  v2f b;
  b.x = base[(k0 + kk) * pitch + n];
  b.y = base[(k0 + kk + 1) * pitch + n];
  return b;
}

__global__ void __launch_bounds__(256, 1)
relattn_fused_kernel(const float* __restrict__ q, const float* __restrict__ k,
                     const float* __restrict__ v, const float* __restrict__ ek,
                     const float* __restrict__ ev, const float* __restrict__ mask,
                     float* __restrict__ out, float* __restrict__ attn) {
  constexpr int T = 1024, D = 64;
  constexpr int SP = 1032;  // score-stripe pitch (floats): 16B-aligned rows, banks step 8/row
  constexpr int QP = 68;    // Q/K/V tile pitch: 16B-aligned rows, banks step 4/row

  extern __shared__ float smem[];
  float* sS     = smem;               // [64][SP]  full 64x1024 score/prob stripe (264 KB)
  float* sQ     = sS + 64 * SP;       // [64][QP]  Q tile
  float* sKV    = sQ + 64 * QP;       // [64][QP]  K or V tile (reused)
  float* sQE    = sKV + 64 * QP;      // [64][34]  qek table, later bucket sums (reused)
  float* sStats = sQE + 64 * 34;      // [128]     row max, row inv-sum

  const int bh   = blockIdx.x >> 4;   // 0..31  (b*H+h)
  const int qt   = blockIdx.x & 15;   // 0..15  query tile
  const int q0   = qt * 64;
  const int tid  = threadIdx.x;
  const int lane = tid & 31;
  const int wave = tid >> 5;          // 8 waves
  const int mi   = wave >> 1;         // 0..3  : 16-row block
  const int ni0  = (wave & 1) * 2;    // 0 or 2: first of two 16-col blocks

  const float* qp = q + ((size_t)bh * T + q0) * D;
  const float* kp = k + (size_t)bh * T * D;
  const float* vp = v + (size_t)bh * T * D;

  // ---- Pass 0: load Q tile (64x64) into LDS ----
  for (int i = tid; i < 1024; i += 256) {
    int r = i >> 4, c4 = (i & 15) << 2;
    *(float4*)(sQ + r * QP + c4) = *(const float4*)(qp + r * 64 + c4);
  }
  __syncthreads();

  // qek[row][rid] = Q[row] . embed_k[rid]   (64x33 tiny GEMM, VALU)
  for (int i = tid; i < 64 * 33; i += 256) {
    int r = i / 33, rid = i - r * 33;
    const float* qr = sQ + r * QP;
    const float* er = ek + rid * 64;
    float acc = 0.f;
#pragma unroll 8
    for (int d0 = 0; d0 < 64; ++d0) acc += qr[d0] * er[d0];
    sQE[r * 34 + rid] = acc;  // consumed after the barrier inside the kt loop
  }

  // ---- Pass 1: logits stripe S[64][1024] = QK^T + rel_k, scaled + masked ----
  for (int kt = 0; kt < 16; ++kt) {
    __syncthreads();  // sKV safe to overwrite (also orders qek writes on kt==0)
    const float* ksrc = kp + (size_t)(kt * 64) * 64;
    for (int i = tid; i < 1024; i += 256) {
      int r = i >> 4, c4 = (i & 15) << 2;
      *(float4*)(sKV + r * QP + c4) = *(const float4*)(ksrc + r * 64 + c4);
    }
    __syncthreads();

    v8f acc0 = {}, acc1 = {};
    const float* Ab = sQ + mi * 16 * QP;
    const float* B0 = sKV + ni0 * 16 * QP;         // K^T frag == row-major frag of K tile
    const float* B1 = sKV + (ni0 + 1) * 16 * QP;
#pragma unroll
    for (int d0 = 0; d0 < 64; d0 += 4) {
      v2f a  = ldfrag_rowmajor(Ab, QP, lane, d0);
      v2f b0 = ldfrag_rowmajor(B0, QP, lane, d0);
      v2f b1 = ldfrag_rowmajor(B1, QP, lane, d0);
      acc0 = WMMA_F32(a, b0, acc0);
      acc1 = WMMA_F32(a, b1, acc1);
    }

    // epilogue: add gathered rel-key term, scale 1/sqrt(64), add mask*-1e9
    const int rofs = mi * 16 + ((lane >> 4) << 3);
    const int n    = lane & 15;
#pragma unroll
    for (int vv = 0; vv < 8; ++vv) {
      int row = rofs + vv;
      int qg  = q0 + row;
      const float* qe = sQE + row * 34;
      {
        int colg = kt * 64 + ni0 * 16 + n;
        int dlt  = min(16, max(-16, colg - qg));
        sS[row * SP + colg] =
            (acc0[vv] + qe[dlt + 16]) * 0.125f + mask[(size_t)qg * T + colg] * (-1e9f);
      }
      {
        int colg = kt * 64 + (ni0 + 1) * 16 + n;
        int dlt  = min(16, max(-16, colg - qg));
        sS[row * SP + colg] =
            (acc1[vv] + qe[dlt + 16]) * 0.125f + mask[(size_t)qg * T + colg] * (-1e9f);
      }
    }
    __syncthreads();
  }

  // ---- Pass 2: row softmax over the resident 64x1024 stripe ----
  const int row2 = tid >> 2, seg = tid & 3;  // 4 threads per row, 256 cols each
  float* rp = sS + row2 * SP + seg * 256;

  float m = -3.4e38f;
#pragma unroll 4
  for (int j = 0; j < 256; j += 4) {
    float4 x = *(const float4*)(rp + j);
    m = fmaxf(m, fmaxf(fmaxf(x.x, x.y), fmaxf(x.z, x.w)));
  }
  sQE[tid] = m;  // qek no longer needed: reuse as reduction scratch
  __syncthreads();
  if (seg == 0)
    sStats[row2] = fmaxf(fmaxf(sQE[tid], sQE[tid + 1]), fmaxf(sQE[tid + 2], sQE[tid + 3]));
  __syncthreads();

  m = sStats[row2];
  float s = 0.f;
#pragma unroll 4
  for (int j = 0; j < 256; j += 4) {
    float4 x = *(float4*)(rp + j);
    x.x = __expf(x.x - m); x.y = __expf(x.y - m);
    x.z = __expf(x.z - m); x.w = __expf(x.w - m);
    s += (x.x + x.y) + (x.z + x.w);
    *(float4*)(rp + j) = x;
  }
  sQE[tid] = s;
  __syncthreads();
  if (seg == 0)
    sStats[64 + row2] = 1.0f / (sQE[tid] + sQE[tid + 1] + sQE[tid + 2] + sQE[tid + 3]);
  __syncthreads();
  for (int i = tid; i < 64 * 34; i += 256) sQE[i] = 0.f;  // now the 33 rel-v buckets
  __syncthreads();

  // normalize in place, stream attn to HBM (written exactly once), accumulate buckets
  {
    const float inv_l = sStats[64 + row2];
    const int qg = q0 + row2;
    float* arow = attn + ((size_t)bh * T + qg) * T + seg * 256;
    float s_lo = 0.f, s_hi = 0.f;
    const int cb = seg * 256;
#pragma unroll 4
    for (int j = 0; j < 256; j += 4) {
      float4 x = *(float4*)(rp + j);
      x.x *= inv_l; x.y *= inv_l; x.z *= inv_l; x.w *= inv_l;
      *(float4*)(rp + j) = x;
      *(float4*)(arow + j) = x;
      float pe[4] = {x.x, x.y, x.z, x.w};
#pragma unroll
      for (int e = 0; e < 4; ++e) {
        int dlt = (cb + j + e) - qg;
        if (dlt <= -16)      s_lo += pe[e];
        else if (dlt >= 16)  s_hi += pe[e];
        else                 atomicAdd(&sQE[row2 * 34 + dlt + 16], pe[e]);  // ds_add_f32
      }
    }
    atomicAdd(&sQE[row2 * 34 + 0],  s_lo);
    atomicAdd(&sQE[row2 * 34 + 32], s_hi);
  }
  __syncthreads();

  // ---- Pass 3: out = P @ V (WMMA over resident stripe) + buckets @ embed_v ----
  v8f o0 = {}, o1 = {};
  for (int kt = 0; kt < 16; ++kt) {
    const float* vsrc = vp + (size_t)(kt * 64) * 64;
    for (int i = tid; i < 1024; i += 256) {
      int r = i >> 4, c4 = (i & 15) << 2;
      *(float4*)(sKV + r * QP + c4) = *(const float4*)(vsrc + r * 64 + c4);
    }
    __syncthreads();

    const float* Ab  = sS + mi * 16 * SP + kt * 64;
    const float* Bb0 = sKV + ni0 * 16;
    const float* Bb1 = sKV + (ni0 + 1) * 16;
#pragma unroll
    for (int c0 = 0; c0 < 64; c0 += 4) {
      v2f a  = ldfrag_rowmajor(Ab, SP, lane, c0);
      v2f b0 = ldfrag_colmajor(Bb0, QP, lane, c0);
      v2f b1 = ldfrag_colmajor(Bb1, QP, lane, c0);
      o0 = WMMA_F32(a, b0, o0);
      o1 = WMMA_F32(a, b1, o1);
    }
    __syncthreads();  // sKV safe to overwrite next iteration
  }

  // rel-value term via bucket sums, then store out
  {
    const int rofs = mi * 16 + ((lane >> 4) << 3);
    const int n    = lane & 15;
#pragma unroll
    for (int vv = 0; vv < 8; ++vv) {
      int row = rofs + vv;
      float a0 = o0[vv], a1 = o1[vv];
      const float* bk = sQE + row * 34;
#pragma unroll
      for (int r = 0; r < 33; ++r) {
        float bw = bk[r];
        a0 += bw * ev[r * 64 + ni0 * 16 + n];
        a1 += bw * ev[r * 64 + (ni0 + 1) * 16 + n];
      }
      size_t ob = ((size_t)bh * T + q0 + row) * D;
      out[ob + ni0 * 16 + n]       = a0;
      out[ob + (ni0 + 1) * 16 + n] = a1;
    }
  }
}

extern "C" void kernel_launch(void* const* d_in, const int* in_sizes, int n_in,
                              void* d_out, int out_size, void* d_ws, size_t ws_size,
                              hipStream_t stream) {
  (void)in_sizes; (void)n_in; (void)d_ws; (void)ws_size; (void)out_size;
  const float* q    = (const float*)d_in[0];
  const float* k    = (const float*)d_in[1];
  const float* v    = (const float*)d_in[2];
  const float* ek   = (const float*)d_in[3];
  const float* ev   = (const float*)d_in[4];
  const float* mask = (const float*)d_in[5];

  float* out  = (float*)d_out;
  float* attn = out + (size_t)2 * 16 * 1024 * 64;  // (out, attn) concatenated

  constexpr size_t lds_bytes =
      (64 * 1032 + 64 * 68 + 64 * 68 + 64 * 34 + 128) * sizeof(float);  // 308224 B <= 320 KB

  (void)hipFuncSetAttribute((const void*)relattn_fused_kernel,
                            hipFuncAttributeMaxDynamicSharedMemorySize, (int)lds_bytes);

  dim3 grid(2 * 16 * (1024 / 64));  // 512 blocks: one per (b,h,64-query stripe)
  dim3 block(256);                  // 8 wave32s
  relattn_fused_kernel<<<grid, block, lds_bytes, stream>>>(q, k, v, ek, ev, mask, out, attn);
}